// AdaptiveFrequencyAsymmetricHuberLoss_14001593385474
// MI455X (gfx1250) — compile-verified
//
#include <hip/hip_runtime.h>

// ---- hyperparameters from the reference ----
#define BASE_DELTA   5.0f
#define MIN_W        1.0f
#define MAX_W        3.0f
#define OVER_W       2.0f
#define UNDER_SCALE  0.05f
// OVER_BIAS == 0.0f

#define NUM_RUL 131
#define BLOCK   256
#define GRID    2048

typedef __attribute__((ext_vector_type(2))) float v2f;
typedef __attribute__((ext_vector_type(8))) float v8f;

// Wave32 all-lane sum using V_WMMA_F32_16X16X4_F32 (see round-1 notes).
// Requires EXEC all-1s: call only from fully converged code.
__device__ __forceinline__ float wave_reduce_wmma(float partial) {
    v2f a; a.x = partial; a.y = 0.0f;
    v2f b; b.x = 1.0f;    b.y = 1.0f;
    v8f c = {};
    c = __builtin_amdgcn_wmma_f32_16x16x4_f32(
            /*neg_a=*/false, a, /*neg_b=*/false, b,
            /*c_mod=*/(short)0, c, /*reuse_a=*/false, /*reuse_b=*/false);
    float s = ((c[0] + c[1]) + (c[2] + c[3])) + ((c[4] + c[5]) + (c[6] + c[7]));
    s += __shfl_xor(s, 16, 32);
    return s;  // every lane holds the wave sum
}

// Per-element math using the precomputed per-RUL table.
// tb = { wf*w_under, wf*w_over, delta, unused }
__device__ __forceinline__ float huber_elem(float p, float t, const float4* stbl) {
    float  e   = p - t;
    float  ae  = fabsf(e);
    int    idx = (int)t;                     // targets are integer-valued RULs
    float4 tb  = stbl[idx];                  // ds_load_b128 gather
    float  w   = (e < 0.0f) ? tb.x : tb.y;
    float  q   = fminf(ae, tb.z);
    float  hub = q * fmaf(-0.5f, q, ae);     // = 0.5e^2 if ae<=d else d*(ae-0.5d)
    return w * hub;
}

// Build the 131-entry table once per block (div + exp run 131x, not 16.7Mx).
__device__ __forceinline__ void build_table(float4* stbl, const float* freq) {
    for (int i = threadIdx.x; i < NUM_RUL; i += BLOCK) {
        float t  = (float)i;
        float wf = fminf(fmaxf(MAX_W / (freq[i] + 1.0f), MIN_W), MAX_W);
        float wu = fmaf(UNDER_SCALE, t, 1.0f);
        float wo = OVER_W * __expf(-t * 0.1f);   // OVER_BIAS == 0
        float d  = BASE_DELTA * fmaf(0.01f, t, 1.0f);
        stbl[i] = make_float4(wf * wu, wf * wo, d, 0.0f);
    }
}

__global__ void __launch_bounds__(BLOCK)
huber_partial_kernel(const float* __restrict__ pred,
                     const float* __restrict__ targ,
                     const float* __restrict__ freq,
                     float* __restrict__ partials,
                     int n4, int n) {
    __shared__ float4 stbl[NUM_RUL];
    __shared__ float  swave[BLOCK / 32];

    build_table(stbl, freq);
    __syncthreads();

    const float4* __restrict__ p4 = (const float4*)pred;
    const float4* __restrict__ t4 = (const float4*)targ;

    float acc = 0.0f;
    const int stride = GRID * BLOCK;
    const int iters  = n4 / stride;          // uniform trip count (8 for N=16.7M)
    int i = blockIdx.x * BLOCK + threadIdx.x;

#pragma unroll 2
    for (int k = 0; k < iters; ++k, i += stride) {
        // gfx1250 global_prefetch_b8 on the next grid-stride tile
        __builtin_prefetch(&p4[i + stride], 0, 0);
        __builtin_prefetch(&t4[i + stride], 0, 0);
        float4 p = p4[i];
        float4 t = t4[i];
        acc += huber_elem(p.x, t.x, stbl);
        acc += huber_elem(p.y, t.y, stbl);
        acc += huber_elem(p.z, t.z, stbl);
        acc += huber_elem(p.w, t.w, stbl);
    }
    // remainder vec4s (none when n4 % stride == 0)
    if (i < n4) {
        float4 p = p4[i];
        float4 t = t4[i];
        acc += huber_elem(p.x, t.x, stbl);
        acc += huber_elem(p.y, t.y, stbl);
        acc += huber_elem(p.z, t.z, stbl);
        acc += huber_elem(p.w, t.w, stbl);
    }
    // scalar tail (empty for N divisible by 4; kept for generality)
    if (blockIdx.x == 0 && threadIdx.x == 0) {
        for (int j = n4 * 4; j < n; ++j)
            acc += huber_elem(pred[j], targ[j], stbl);
    }

    // fully converged here: EXEC all-1s for WMMA
    float wsum = wave_reduce_wmma(acc);
    const int lane = threadIdx.x & 31;
    const int wid  = threadIdx.x >> 5;
    if (lane == 0) swave[wid] = wsum;
    __syncthreads();
    if (threadIdx.x == 0) {
        float s = 0.0f;
#pragma unroll
        for (int w = 0; w < BLOCK / 32; ++w) s += swave[w];
        partials[blockIdx.x] = s;
    }
}

__global__ void __launch_bounds__(BLOCK)
huber_final_kernel(const float* __restrict__ partials,
                   float* __restrict__ out, float inv_n) {
    __shared__ float swave[BLOCK / 32];
    float acc = 0.0f;
    for (int i = threadIdx.x; i < GRID; i += BLOCK) acc += partials[i];

    float wsum = wave_reduce_wmma(acc);
    const int lane = threadIdx.x & 31;
    const int wid  = threadIdx.x >> 5;
    if (lane == 0) swave[wid] = wsum;
    __syncthreads();
    if (threadIdx.x == 0) {
        float s = 0.0f;
#pragma unroll
        for (int w = 0; w < BLOCK / 32; ++w) s += swave[w];
        out[0] = s * inv_n;
    }
}

extern "C" void kernel_launch(void* const* d_in, const int* in_sizes, int n_in,
                              void* d_out, int out_size, void* d_ws, size_t ws_size,
                              hipStream_t stream) {
    const float* pred = (const float*)d_in[0];
    const float* targ = (const float*)d_in[1];
    const float* freq = (const float*)d_in[2];
    float* out   = (float*)d_out;
    float* parts = (float*)d_ws;   // GRID floats = 8 KB of scratch

    const int n  = in_sizes[0];
    const int n4 = n / 4;

    huber_partial_kernel<<<GRID, BLOCK, 0, stream>>>(pred, targ, freq, parts, n4, n);
    huber_final_kernel<<<1, BLOCK, 0, stream>>>(parts, out, 1.0f / (float)n);
}